// ChamferBoundarySDFLossVec_60189671686289
// MI455X (gfx1250) — compile-verified
//
#include <hip/hip_runtime.h>
#include <hip/hip_bf16.h>

// ---------------- problem constants (match reference) ----------------
constexpr int   H = 768;
constexpr int   W = 768;
constexpr int   B = 4;
constexpr int   K = 4096;
constexpr float EPSF = 1e-8f;

constexpr int NZ    = H * W;            // exact-zero candidates
constexpr int NV    = (H - 1) * W;      // vertical-crossing candidates
constexpr int NHC   = H * (W - 1);      // horizontal-crossing candidates
constexpr int NCAND = NZ + NV + NHC;
constexpr int SCAN_T = 256;
constexpr int NBLK   = (NCAND + SCAN_T - 1) / SCAN_T;
constexpr int NIMG   = 2 * B;           // pred + gt per batch

constexpr int WAVES   = 4;              // waves per chamfer workgroup
constexpr int CTHREADS = 32 * WAVES;    // 128 threads
constexpr int CHUNK   = 2048;           // gt points staged in LDS per round (32 KB)

typedef __attribute__((ext_vector_type(2))) float v2f;
typedef __attribute__((ext_vector_type(8))) float v8f;

// ---------------- candidate decode (exactly the reference's concat order) ----
__device__ __forceinline__ bool decode_cand(const float* __restrict__ s, int cand,
                                            float& pr, float& pc) {
  if (cand < NZ) {
    int r = cand / W, c = cand - r * W;
    pr = (float)r; pc = (float)c;
    return s[cand] == 0.0f;
  } else if (cand < NZ + NV) {
    int i = cand - NZ;
    int r = i / W, c = i - r * W;
    float v1 = s[r * W + c], v2 = s[(r + 1) * W + c];
    float a = fabsf(v1) / (fabsf(v1) + fabsf(v2) + EPSF);
    pr = (float)r + a; pc = (float)c;
    return (v1 * v2) < 0.0f;
  } else {
    int i = cand - NZ - NV;
    int r = i / (W - 1), c = i - r * (W - 1);
    float h1 = s[r * W + c], h2 = s[r * W + c + 1];
    float a = fabsf(h1) / (fabsf(h1) + fabsf(h2) + EPSF);
    pr = (float)r; pc = (float)c + a;
    return (h1 * h2) < 0.0f;
  }
}

__device__ __forceinline__ void corner(float r, float c, int& r0, int& c0,
                                       int& r1, int& c1, float& ar, float& ac) {
  r0 = (int)floorf(r); r0 = min(max(r0, 0), H - 1);
  c0 = (int)floorf(c); c0 = min(max(c0, 0), W - 1);
  r1 = min(r0 + 1, H - 1);
  c1 = min(c0 + 1, W - 1);
  ar = r - (float)r0;
  ac = c - (float)c0;
}

__device__ __forceinline__ void normal_at(const float* __restrict__ s, int r, int c,
                                          float& gr, float& gc) {
  if (r == 0)          gr = s[W + c] - s[c];
  else if (r == H - 1) gr = s[r * W + c] - s[(r - 1) * W + c];
  else                 gr = 0.5f * (s[(r + 1) * W + c] - s[(r - 1) * W + c]);
  if (c == 0)          gc = s[r * W + 1] - s[r * W];
  else if (c == W - 1) gc = s[r * W + c] - s[r * W + c - 1];
  else                 gc = 0.5f * (s[r * W + c + 1] - s[r * W + c - 1]);
}

// ---------------- pass 1: per-block valid counts ----------------
__global__ void count_kernel(const float* __restrict__ pred,
                             const float* __restrict__ gt,
                             int* __restrict__ bc) {
  int img = blockIdx.y;
  const float* base = (img & 1) ? gt : pred;
  const float* s = base + (size_t)(img >> 1) * H * W;
  int cand = blockIdx.x * SCAN_T + threadIdx.x;
  bool msk = false;
  float pr, pc;
  if (cand < NCAND) msk = decode_cand(s, cand, pr, pc);
  __shared__ int cnt;
  if (threadIdx.x == 0) cnt = 0;
  __syncthreads();
  if (msk) atomicAdd(&cnt, 1);
  __syncthreads();
  if (threadIdx.x == 0) bc[(size_t)img * NBLK + blockIdx.x] = cnt;
}

// ---------------- pass 2: serial exclusive scan per image ----------------
__global__ void scan_kernel(const int* __restrict__ bc, int* __restrict__ bo,
                            int* __restrict__ totals) {
  int img = blockIdx.x;
  if (threadIdx.x != 0) return;
  const int* in = bc + (size_t)img * NBLK;
  int* outp = bo + (size_t)img * NBLK;
  int run = 0;
  for (int i = 0; i < NBLK; ++i) { outp[i] = run; run += in[i]; }
  totals[img] = run < K ? run : K;   // argsort(~mask)[:K] truncation
}

// ---------------- pass 3: ordered compaction of zc points ----------------
__global__ void write_kernel(const float* __restrict__ pred,
                             const float* __restrict__ gt,
                             const int* __restrict__ bo,
                             float* __restrict__ pts) {
  int img = blockIdx.y;
  const float* base = (img & 1) ? gt : pred;
  const float* s = base + (size_t)(img >> 1) * H * W;
  int tid = threadIdx.x;
  int cand = blockIdx.x * SCAN_T + tid;
  bool msk = false;
  float pr = 0.f, pc = 0.f;
  if (cand < NCAND) msk = decode_cand(s, cand, pr, pc);
  __shared__ int sc[SCAN_T];
  sc[tid] = msk ? 1 : 0;
  __syncthreads();
  for (int off = 1; off < SCAN_T; off <<= 1) {
    int v = (tid >= off) ? sc[tid - off] : 0;
    __syncthreads();
    sc[tid] += v;
    __syncthreads();
  }
  if (msk) {
    int rank = sc[tid] - 1;  // exclusive rank among masked in this block
    int pos = bo[(size_t)img * NBLK + blockIdx.x] + rank;
    if (pos < K) {
      float* dst = pts + ((size_t)img * K + pos) * 2;
      dst[0] = pr; dst[1] = pc;
    }
  }
}

__global__ void init_out(float* out) { out[0] = 0.0f; }

// ---------------- pass 4: WMMA chamfer + argmin + loss terms ----------------
// 4 waves per workgroup; each wave owns 16 pred points. gt B-pack
// [-2gr, -2gc, 1, |g|^2] is staged into LDS in 2048-point chunks and reused by
// all 4 waves; the WMMA inner loop feeds from ds_load_b64.
//   A row  = [pr, pc, |p|^2, 1]            (16x4, f32)
//   B col  = [-2gr, -2gc, 1, |g|^2]        (4x16, f32)
//   D[m][n] = |p_m - g_n|^2                (16x16, f32)
__global__ void __launch_bounds__(CTHREADS)
chamfer_kernel(const float* __restrict__ pred_all,
               const float* __restrict__ pts,
               const int* __restrict__ totals,
               float* __restrict__ out) {
  int b = blockIdx.y;
  const float* s    = pred_all + (size_t)b * H * W;
  const float* ppts = pts + (size_t)(2 * b + 0) * K * 2;
  const float* gpts = pts + (size_t)(2 * b + 1) * K * 2;
  int np = totals[2 * b + 0];
  int ng = totals[2 * b + 1];

  int tid  = threadIdx.x;
  int wave = tid >> 5;
  int lane = tid & 31;
  int jl   = lane & 15;
  bool hi  = lane >= 16;

  int mbase = blockIdx.x * (16 * WAVES) + wave * 16;
  bool active = (mbase < np);

  __shared__ float4 sB[CHUNK];         // staged gt B-pack (32 KB)
  __shared__ float  sbd[WAVES][16];
  __shared__ int    sbi[WAVES][16];

  // A-matrix: lanes 0-15 hold K=0,1 (pr,pc); lanes 16-31 hold K=2,3 (|p|^2, 1)
  float pr = 0.f, pc = 0.f;
  int m = mbase + jl;
  if (active && m < np) { pr = ppts[2 * m]; pc = ppts[2 * m + 1]; }
  v2f a;
  a.x = hi ? (pr * pr + pc * pc) : pr;
  a.y = hi ? 1.0f : pc;

  float bd[8];
  int   bi[8];
#pragma unroll
  for (int v = 0; v < 8; ++v) { bd[v] = 3.0e38f; bi[v] = 0; }

  int nchunks = (ng + CHUNK - 1) / CHUNK;   // uniform across block
  for (int ch = 0; ch < nchunks; ++ch) {
    int base   = ch * CHUNK;
    int cnt    = min(CHUNK, ng - base);
    int cntPad = (cnt + 15) & ~15;

    __syncthreads();                        // previous chunk fully consumed
    for (int idx = tid; idx < cntPad; idx += CTHREADS) {
      float4 pk;
      if (idx < cnt) {
        float gr = gpts[2 * (base + idx)], gc = gpts[2 * (base + idx) + 1];
        pk = make_float4(-2.0f * gr, -2.0f * gc, 1.0f, gr * gr + gc * gc);
      } else {
        pk = make_float4(0.0f, 0.0f, 1.0f, 1.0e30f);  // pad -> BIG distance
      }
      sB[idx] = pk;
    }
    __syncthreads();                        // chunk staged

    if (active) {
      int nt = cntPad >> 4;
      const float* sBf = (const float*)sB;
#pragma unroll 4
      for (int t = 0; t < nt; ++t) {
        int j = t * 16 + jl;
        // lanes 0-15: K=0,1 ; lanes 16-31: K=2,3 of column j
        v2f bb = *(const v2f*)(sBf + 4 * j + (hi ? 2 : 0));
        v8f cacc = {};
        cacc = __builtin_amdgcn_wmma_f32_16x16x4_f32(false, a, false, bb,
                                                     (short)0, cacc, false, false);
#pragma unroll
        for (int v = 0; v < 8; ++v) {
          float d = cacc[v];
          int idx = base + t * 16 + jl;
          if (d < bd[v]) { bd[v] = d; bi[v] = idx; }
        }
      }
    }
  }

  // argmin across the 16 lanes of each half (xor masks 1,2,4,8 stay in-half)
  if (active) {
#pragma unroll
    for (int v = 0; v < 8; ++v) {
#pragma unroll
      for (int off = 1; off < 16; off <<= 1) {
        float od = __shfl_xor(bd[v], off, 32);
        int   oi = __shfl_xor(bi[v], off, 32);
        if (od < bd[v] || (od == bd[v] && oi < bi[v])) { bd[v] = od; bi[v] = oi; }
      }
    }
    if (jl == 0) {
#pragma unroll
      for (int v = 0; v < 8; ++v) {
        int row = v + (hi ? 8 : 0);
        sbd[wave][row] = bd[v];
        sbi[wave][row] = bi[v];
      }
    }
  }
  __syncthreads();

  // per-pred-point tail: normals, dot, bilinear sample, loss terms
  float contrib = 0.0f;
  int mm = mbase + lane;
  if (active && lane < 16 && mm < np) {
    float r = ppts[2 * mm], c = ppts[2 * mm + 1];
    int r0, c0, r1, c1; float ar, ac;
    corner(r, c, r0, c0, r1, c1, ar, ac);
    float w00 = (1.f - ar) * (1.f - ac), w01 = (1.f - ar) * ac;
    float w10 = ar * (1.f - ac),         w11 = ar * ac;
    float nr00, nc00, nr01, nc01, nr10, nc10, nr11, nc11;
    normal_at(s, r0, c0, nr00, nc00);
    normal_at(s, r0, c1, nr01, nc01);
    normal_at(s, r1, c0, nr10, nc10);
    normal_at(s, r1, c1, nr11, nc11);
    float nr = w00 * nr00 + w01 * nr01 + w10 * nr10 + w11 * nr11;
    float nc = w00 * nc00 + w01 * nc01 + w10 * nc10 + w11 * nc11;
    float inv = 1.0f / (sqrtf(nr * nr + nc * nc) + 1e-8f);
    nr *= inv; nc *= inv;
    float bil = w00 * s[r0 * W + c0] + w01 * s[r0 * W + c1] +
                w10 * s[r1 * W + c0] + w11 * s[r1 * W + c1];
    float d2 = sbd[wave][lane];
    float dotv = 0.0f;
    if (ng > 0 && d2 <= 9.0f) {            // DIST_THRESHOLD^2, UPDATE_SCALE=1
      int gi = sbi[wave][lane];
      float gr = gpts[2 * gi], gc = gpts[2 * gi + 1];
      dotv = (gr - r) * nr + (gc - c) * nc;
    }
    contrib = dotv * bil + bil;            // W_INJECT*inject + W_PIXEL*pixel
  }
#pragma unroll
  for (int off = 16; off > 0; off >>= 1) contrib += __shfl_down(contrib, off, 32);
  if (lane == 0 && active) atomicAdd(out, contrib * (1.0f / (float)B));
}

// ---------------- launch ----------------
extern "C" void kernel_launch(void* const* d_in, const int* in_sizes, int n_in,
                              void* d_out, int out_size, void* d_ws, size_t ws_size,
                              hipStream_t stream) {
  (void)in_sizes; (void)n_in; (void)out_size; (void)ws_size;
  const float* pred = (const float*)d_in[0];
  const float* gt   = (const float*)d_in[1];
  float* out = (float*)d_out;

  int*   bc     = (int*)d_ws;                       // [NIMG][NBLK]
  int*   bo     = bc + (size_t)NIMG * NBLK;         // [NIMG][NBLK]
  int*   totals = bo + (size_t)NIMG * NBLK;         // [NIMG]
  float* pts    = (float*)(totals + NIMG);          // [NIMG][K][2]

  init_out<<<1, 1, 0, stream>>>(out);
  count_kernel<<<dim3(NBLK, NIMG), SCAN_T, 0, stream>>>(pred, gt, bc);
  scan_kernel<<<dim3(NIMG), 32, 0, stream>>>(bc, bo, totals);
  write_kernel<<<dim3(NBLK, NIMG), SCAN_T, 0, stream>>>(pred, gt, bo, pts);
  chamfer_kernel<<<dim3(K / (16 * WAVES), B), CTHREADS, 0, stream>>>(pred, pts,
                                                                     totals, out);
}